// HungarianMatcher_54434415509808
// MI455X (gfx1250) — compile-verified
//
#include <hip/hip_runtime.h>
#include <hip/hip_bf16.h>
#include <stdint.h>

// ---------------------------------------------------------------------------
// Hungarian matcher (linear sum assignment), B=32 batches of a 128x128 padded
// LSAP. One single-wave (wave32) workgroup per batch:
//   - lane j owns columns 4j..4j+3  (duals v, minv, used live in VGPRs)
//   - cost matrix (128x96, 48KB) + u/p/way arrays live in LDS (CDNA5: 320KB/WGP)
//   - per-step argmin = 5x shfl_xor butterfly, no barriers needed (1 wave)
//   - inputs staged via CDNA5 async global->LDS (ASYNCcnt) when available
// ---------------------------------------------------------------------------

namespace {
constexpr int kB = 32;    // batches
constexpr int kP = 128;   // predictions (rows)
constexpr int kT = 96;    // targets (stored cols; cols >= T are always 0)
constexpr int kN = 128;   // padded square LSAP size
}

// Pointer types matching the async-to-LDS builtin signatures discovered from
// the round-1 diagnostics:
//   b128: (v4i32 AS(1)*, lds AS(3)*, imm offset, imm cpol)
//   b32 : (i32   AS(1)*, lds AS(3)*, imm offset, imm cpol)
typedef int v4i_t __attribute__((vector_size(16)));
typedef __attribute__((address_space(1))) v4i_t* gas_v4i;
typedef __attribute__((address_space(1))) int*   gas_i32;
typedef __attribute__((address_space(3))) v4i_t* las_v4i;
typedef __attribute__((address_space(3))) int*   las_i32;

#if defined(__AMDGCN__)
 #if __has_builtin(__builtin_amdgcn_global_load_async_to_lds_b128) && \
     __has_builtin(__builtin_amdgcn_global_load_async_to_lds_b32)  && \
     __has_builtin(__builtin_amdgcn_s_wait_asynccnt)
  #define HM_ASYNC_LDS 1
 #endif
#endif

__device__ __forceinline__ void wave_argmin(float& v, int& j) {
#pragma unroll
  for (int off = 16; off > 0; off >>= 1) {
    float ov = __shfl_xor(v, off, 32);
    int   oj = __shfl_xor(j, off, 32);
    if (ov < v || (ov == v && oj < j)) { v = ov; j = oj; }  // ties -> lowest col
  }
}

__device__ __forceinline__ int wave_sum_i32(int x) {
#pragma unroll
  for (int off = 16; off > 0; off >>= 1) x += __shfl_xor(x, off, 32);
  return x;
}

__global__ __launch_bounds__(32)
void hungarian_lsap_kernel(const float* __restrict__ pred_mz,
                           const float* __restrict__ pred_in,
                           const float* __restrict__ pred_cf,
                           const float* __restrict__ tgt_mz,
                           const float* __restrict__ tgt_in,
                           const int*  __restrict__ tgt_mask,
                           float* __restrict__ out)
{
  __shared__ __align__(16) float s_cost[kP * kT];   // 48 KB, row stride kT
  __shared__ __align__(16) float s_pmz[kP];
  __shared__ __align__(16) float s_pin[kP];
  __shared__ __align__(16) float s_pcf[kP];
  __shared__ __align__(16) float s_tmz[kT];
  __shared__ __align__(16) float s_tin[kT];
  __shared__ float s_u[kN + 4];
  __shared__ int   s_p[kN + 4];
  __shared__ int   s_way[kN + 4];

  const int b    = blockIdx.x;
  const int lane = threadIdx.x;

#if __has_builtin(__builtin_amdgcn_s_wait_tensorcnt)
  __builtin_amdgcn_s_wait_tensorcnt(0);   // CDNA5 split dep-counter path (idle here)
#endif

  // L2 prefetch of this batch's slices (lowers to global_prefetch_b8)
  __builtin_prefetch(pred_mz + b * kP, 0, 1);
  __builtin_prefetch(pred_in + b * kP, 0, 1);
  __builtin_prefetch(tgt_mz  + b * kT, 0, 1);
  __builtin_prefetch(tgt_in  + b * kT, 0, 1);

  // ---- stage per-batch vectors into LDS (async on CDNA5) ---------------
#ifdef HM_ASYNC_LDS
  {
    __builtin_amdgcn_global_load_async_to_lds_b128(
        (gas_v4i)(pred_mz + b * kP + lane * 4), (las_v4i)&s_pmz[lane * 4], 0, 0);
    __builtin_amdgcn_global_load_async_to_lds_b128(
        (gas_v4i)(pred_in + b * kP + lane * 4), (las_v4i)&s_pin[lane * 4], 0, 0);
    __builtin_amdgcn_global_load_async_to_lds_b128(
        (gas_v4i)(pred_cf + b * kP + lane * 4), (las_v4i)&s_pcf[lane * 4], 0, 0);
#pragma unroll
    for (int t = 0; t < 3; ++t) {
      const int k = t * 32 + lane;
      __builtin_amdgcn_global_load_async_to_lds_b32(
          (gas_i32)(tgt_mz + b * kT + k), (las_i32)&s_tmz[k], 0, 0);
      __builtin_amdgcn_global_load_async_to_lds_b32(
          (gas_i32)(tgt_in + b * kT + k), (las_i32)&s_tin[k], 0, 0);
    }
  }
#else
  for (int k = lane; k < kP; k += 32) {
    s_pmz[k] = pred_mz[b * kP + k];
    s_pin[k] = pred_in[b * kP + k];
    s_pcf[k] = pred_cf[b * kP + k];
  }
  for (int k = lane; k < kT; k += 32) {
    s_tmz[k] = tgt_mz[b * kT + k];
    s_tin[k] = tgt_in[b * kT + k];
  }
#endif

  // valid-target count m (overlaps async staging; reads globals directly)
  int m = 0;
  for (int k = lane; k < kT; k += 32) m += tgt_mask[b * kT + k];
  m = wave_sum_i32(m);

  for (int k = lane; k < kN + 4; k += 32) { s_u[k] = 0.0f; s_p[k] = 0; s_way[k] = 0; }

#ifdef HM_ASYNC_LDS
  __builtin_amdgcn_s_wait_asynccnt(0);
#endif
  __syncthreads();

  // ---- build zero-padded cost matrix in LDS ----------------------------
  // cost[i][j] = |pmz_i - tmz_j| + |pin_i - tin_j| - pcf_i   (j < m, else 0)
  const bool colOwner = (lane < kT / 4);    // lanes 0..23 own stored columns
  float tm[4], ti[4];
#pragma unroll
  for (int c = 0; c < 4; ++c) {
    const int col = lane * 4 + c;
    tm[c] = colOwner ? s_tmz[col] : 0.0f;
    ti[c] = colOwner ? s_tin[col] : 0.0f;
  }
  for (int i = 0; i < kP; ++i) {
    const float pm = s_pmz[i], pi = s_pin[i], pc = s_pcf[i];
    if (colOwner) {
      float4 r;
      float* rr = (float*)&r;
#pragma unroll
      for (int c = 0; c < 4; ++c) {
        const int col = lane * 4 + c;
        rr[c] = (col < m) ? (fabsf(pm - tm[c]) + fabsf(pi - ti[c]) - pc) : 0.0f;
      }
      *(float4*)&s_cost[i * kT + lane * 4] = r;
    }
  }
  __syncthreads();

  // ---- Hungarian: shortest augmenting path with dual potentials --------
  const float INF = __builtin_inff();
  float vdual[4];                     // v[j] for this lane's 4 columns
#pragma unroll
  for (int c = 0; c < 4; ++c) vdual[c] = 0.0f;

  for (int i = 1; i <= kN; ++i) {
    if (lane == 0) s_p[0] = i;
    __syncthreads();

    float minv[4];
#pragma unroll
    for (int c = 0; c < 4; ++c) minv[c] = INF;
    int usedm = 0;                    // 4-bit used mask for this lane's cols
    int j0 = 0;

    while (true) {
      // used[j0] = true (owner lane flips its register bit)
      if (j0 > 0) {
        const int ci = j0 - 1;
        if ((ci >> 2) == lane) usedm |= (1 << (ci & 3));
      }
      const int   i0  = s_p[j0];      // uniform LDS broadcast read
      const float ui0 = s_u[i0];

      float4 crow;
      if (colOwner)
        crow = *(const float4*)&s_cost[(i0 - 1) * kT + lane * 4];
      else
        crow = make_float4(0.f, 0.f, 0.f, 0.f);   // cols 96..127: pure padding
      const float* cr = (const float*)&crow;

      // relax edges out of row i0; fold local argmin over unused columns
      float bestv = INF;
      int   bestj = kN + 1;
#pragma unroll
      for (int c = 0; c < 4; ++c) {
        const int col = lane * 4 + c;
        if (!((usedm >> c) & 1)) {
          const float cur = cr[c] - ui0 - vdual[c];
          if (cur < minv[c]) { minv[c] = cur; s_way[col + 1] = j0; }
          if (minv[c] < bestv) { bestv = minv[c]; bestj = col + 1; }
        }
      }
      wave_argmin(bestv, bestj);      // uniform (delta, j1) on every lane
      const float delta = bestv;
      const int   j1    = bestj;

      // dual updates: u[p[used]] += d, v[used] -= d, minv[~used] -= d
#pragma unroll
      for (int c = 0; c < 4; ++c) {
        const int col = lane * 4 + c;
        if ((usedm >> c) & 1) {
          const int pj = s_p[col + 1];   // distinct rows -> race-free LDS RMW
          s_u[pj] += delta;
          vdual[c] -= delta;
        } else {
          minv[c] -= delta;
        }
      }
      if (lane == 0) s_u[i] += delta;    // virtual column 0 carries row i

      j0 = j1;
      if (s_p[j0] == 0) break;           // reached a free column
    }
    __syncthreads();

    // augment along the alternating path (serial, lane 0)
    if (lane == 0) {
      while (j0 != 0) {
        const int j1 = s_way[j0];
        s_p[j0] = s_p[j1];
        j0 = j1;
      }
    }
    __syncthreads();
  }

  // ---- emit (pred_idx, tgt_idx, counts), sorted by assigned row --------
  float* out_pred = out;
  float* out_tgt  = out + kB * kT;
  float* out_cnt  = out + 2 * kB * kT;

  for (int k = lane; k < kT; k += 32) {
    if (k < m) {
      const int rk = s_p[k + 1] - 1;     // row assigned to column k
      int rank = 0;
      for (int l = 0; l < m; ++l) rank += (s_p[l + 1] - 1 < rk) ? 1 : 0;
      out_pred[b * kT + rank] = (float)rk;
      out_tgt [b * kT + rank] = (float)k;
    } else {
      out_pred[b * kT + k] = -1.0f;
      out_tgt [b * kT + k] = -1.0f;
    }
  }
  if (lane == 0) out_cnt[b] = (float)m;
}

extern "C" void kernel_launch(void* const* d_in, const int* in_sizes, int n_in,
                              void* d_out, int out_size, void* d_ws, size_t ws_size,
                              hipStream_t stream) {
  (void)in_sizes; (void)n_in; (void)out_size; (void)d_ws; (void)ws_size;
  const float* pred_mz = (const float*)d_in[0];
  const float* pred_in = (const float*)d_in[1];
  const float* pred_cf = (const float*)d_in[2];
  const float* tgt_mz  = (const float*)d_in[3];
  const float* tgt_in  = (const float*)d_in[4];
  const int*   tmask   = (const int*)d_in[5];

  hipLaunchKernelGGL(hungarian_lsap_kernel, dim3(kB), dim3(32), 0, stream,
                     pred_mz, pred_in, pred_cf, tgt_mz, tgt_in, tmask,
                     (float*)d_out);
}